// Dihedral_78950088835407
// MI455X (gfx1250) — compile-verified
//
#include <hip/hip_runtime.h>
#include <stdint.h>

static constexpr int kDih   = 2000000;
static constexpr int kTypes = 25;
static constexpr int TILE   = 256;

// ---------------- CDNA5 async global->LDS helpers (gfx1250) ----------------
// LDS byte offset passed in a 32-bit VGPR; global address passed as a 64-bit VGPR pair.
__device__ __forceinline__ void async_load_b32(uint32_t lds_off, const void* gaddr) {
  asm volatile("global_load_async_to_lds_b32 %0, %1, off"
               :: "v"(lds_off), "v"((unsigned long long)(uintptr_t)gaddr)
               : "memory");
}
__device__ __forceinline__ void s_wait_asynccnt_0() { asm volatile("s_wait_asynccnt 0x0" ::: "memory"); }
__device__ __forceinline__ void s_wait_asynccnt_5() { asm volatile("s_wait_asynccnt 0x5" ::: "memory"); }
__device__ __forceinline__ void s_wait_dscnt_0()    { asm volatile("s_wait_dscnt 0x0"    ::: "memory"); }

__global__ void zero_f32(float* p, int n) {
  int i = blockIdx.x * blockDim.x + threadIdx.x;
  if (i < n) p[i] = 0.0f;
}

__global__ __launch_bounds__(TILE)
void dihedral_energy(const float* __restrict__ pos,
                     const float* __restrict__ th0, const float* __restrict__ kk0,
                     const float* __restrict__ th1, const float* __restrict__ kk1,
                     const float* __restrict__ th2, const float* __restrict__ kk2,
                     const int*  __restrict__ mapping,   // [4, kDih]
                     const int*  __restrict__ atype,     // [N_ATOMS]
                     const int*  __restrict__ mbatch,    // [kDih], sorted
                     float* __restrict__ out)            // [N_MOLECULES]
{
  __shared__ int stage[2][5][TILE];   // 10 KB: idx0..3 + molecule id, double-buffered
  const int tid  = threadIdx.x;
  const int lane = tid & 31;
  const int ntiles = (kDih + TILE - 1) / TILE;

  // Issue the 5 async streaming loads for tile t into buffer b (clamped => uniform EXEC).
  auto issue = [&](int t, int b) {
    long long gi = (long long)t * TILE + tid;
    int j = (gi < (long long)kDih) ? (int)gi : (kDih - 1);
    async_load_b32((uint32_t)(uintptr_t)&stage[b][0][tid], mapping + (size_t)0 * kDih + j);
    async_load_b32((uint32_t)(uintptr_t)&stage[b][1][tid], mapping + (size_t)1 * kDih + j);
    async_load_b32((uint32_t)(uintptr_t)&stage[b][2][tid], mapping + (size_t)2 * kDih + j);
    async_load_b32((uint32_t)(uintptr_t)&stage[b][3][tid], mapping + (size_t)3 * kDih + j);
    async_load_b32((uint32_t)(uintptr_t)&stage[b][4][tid], mbatch + j);
  };

  int buf = 0;
  const int t0 = (int)blockIdx.x;
  if (t0 < ntiles) issue(t0, 0);

  for (int t = t0; t < ntiles; t += (int)gridDim.x) {
    const int  tn   = t + (int)gridDim.x;
    const bool more = tn < ntiles;
    if (more) {
      s_wait_dscnt_0();        // prior LDS reads of the buffer we are about to overwrite
      issue(tn, buf ^ 1);
      // prefetch two tiles ahead on the streaming rows (global_prefetch_b8)
      long long g2 = (long long)(tn + (int)gridDim.x) * TILE + tid;
      if (g2 < (long long)kDih) {
        __builtin_prefetch(mapping + (size_t)0 * kDih + (size_t)g2, 0, 0);
        __builtin_prefetch(mapping + (size_t)3 * kDih + (size_t)g2, 0, 0);
      }
      s_wait_asynccnt_5();     // the 5 loads of tile t (older) are complete, newest 5 in flight
    } else {
      s_wait_asynccnt_0();
    }

    const int i0  = stage[buf][0][tid];
    const int i1  = stage[buf][1][tid];
    const int i2  = stage[buf][2][tid];
    const int i3  = stage[buf][3][tid];
    const int seg = stage[buf][4][tid];

    const long long gi = (long long)t * TILE + tid;
    const bool valid = gi < (long long)kDih;

    // position gathers (L2-resident, 12B each; compiler merges to b96)
    const float* p0 = pos + 3 * (size_t)i0;
    const float* p1 = pos + 3 * (size_t)i1;
    const float* p2 = pos + 3 * (size_t)i2;
    const float* p3 = pos + 3 * (size_t)i3;
    float p0x = p0[0], p0y = p0[1], p0z = p0[2];
    float p1x = p1[0], p1y = p1[1], p1z = p1[2];
    float p2x = p2[0], p2y = p2[1], p2z = p2[2];
    float p3x = p3[0], p3y = p3[1], p3z = p3[2];

    float b1x = p1x - p0x, b1y = p1y - p0y, b1z = p1z - p0z;
    float b2x = p2x - p1x, b2y = p2y - p1y, b2z = p2z - p1z;
    float b3x = p3x - p2x, b3y = p3y - p2y, b3z = p3z - p2z;

    float n1x = b1y * b2z - b1z * b2y;
    float n1y = b1z * b2x - b1x * b2z;
    float n1z = b1x * b2y - b1y * b2x;
    float n2x = b2y * b3z - b2z * b3y;
    float n2y = b2z * b3x - b2x * b3z;
    float n2z = b2x * b3y - b2y * b3x;

    float xdot = n1x * n2x + n1y * n2y + n1z * n2z;
    float cxx = n1y * n2z - n1z * n2y;
    float cxy = n1z * n2x - n1x * n2z;
    float cxz = n1x * n2y - n1y * n2x;
    float b2n = sqrtf(b2x * b2x + b2y * b2y + b2z * b2z);
    float ydot = (cxx * b2x + cxy * b2y + cxz * b2z) / b2n;
    float theta = atan2f(ydot, xdot);

    // parameter-table gather (single shared index into six 9.4MB-total tables)
    int a0 = atype[i0], a1 = atype[i1], a2 = atype[i2], a3 = atype[i3];
    int pidx = ((a0 * kTypes + a1) * kTypes + a2) * kTypes + a3;

    float V = kk0[pidx] * (1.0f - cosf(theta - th0[pidx]))
            + kk1[pidx] * (1.0f - cosf(2.0f * theta - th1[pidx]))
            + kk2[pidx] * (1.0f - cosf(3.0f * theta - th2[pidx]));
    float v = valid ? V : 0.0f;

    // wave32 segmented inclusive scan over sorted molecule ids
#pragma unroll
    for (int off = 1; off < 32; off <<= 1) {
      float vu = __shfl_up(v, off, 32);
      int   su = __shfl_up(seg, off, 32);
      if (lane >= off && su == seg) v += vu;
    }
    int sdn = __shfl_down(seg, 1, 32);
    if (lane == 31 || sdn != seg)
      atomicAdd(&out[seg], v);   // ~1-2 f32 atomics per wave into L2-resident bins

    buf ^= 1;
  }
}

extern "C" void kernel_launch(void* const* d_in, const int* in_sizes, int n_in,
                              void* d_out, int out_size, void* d_ws, size_t ws_size,
                              hipStream_t stream) {
  (void)in_sizes; (void)n_in; (void)d_ws; (void)ws_size;
  const float* pos = (const float*)d_in[0];
  const float* th0 = (const float*)d_in[1];
  const float* k0  = (const float*)d_in[2];
  const float* th1 = (const float*)d_in[3];
  const float* k1  = (const float*)d_in[4];
  const float* th2 = (const float*)d_in[5];
  const float* k2  = (const float*)d_in[6];
  const int* mapping = (const int*)d_in[7];
  const int* atype   = (const int*)d_in[8];
  const int* mbatch  = (const int*)d_in[9];
  float* out = (float*)d_out;

  zero_f32<<<(out_size + 255) / 256, 256, 0, stream>>>(out, out_size);

  const int ntiles = (kDih + TILE - 1) / TILE;
  const int blocks = ntiles < 1024 ? ntiles : 1024;   // ~7.6 tiles per block -> pipeline pays off
  dihedral_energy<<<blocks, TILE, 0, stream>>>(pos, th0, k0, th1, k1, th2, k2,
                                               mapping, atype, mbatch, out);
}